// Hahow_Loss_22110491639970
// MI455X (gfx1250) — compile-verified
//
#include <hip/hip_runtime.h>
#include <stdint.h>

// MAP@K (K=50) scorer for MI455X / gfx1250.
// Memory-bound (~13-26 MB total): staged via GLOBAL_LOAD_ASYNC_TO_LDS (CDNA5
// async path, ASYNCcnt), computed with wave32 ballot/popcount prefix logic.
// Deterministic two-stage reduction (per-block partials -> fixed-order sum).

#define KTOP 50
#define PRED_COLS 100
#define ROWS_PER_BLOCK 8   // 8 waves * 32 lanes = 256 threads, one wave per row

// Low 32 bits of a generic (flat) pointer to __shared__ data == LDS byte
// offset within the workgroup allocation (ISA 10.2).
__device__ __forceinline__ uint32_t lds_off(const void* p) {
    return (uint32_t)(uintptr_t)p;
}

template <bool USE_ATOMIC>
__global__ __launch_bounds__(256) void mapk_kernel(
    const unsigned long long* __restrict__ y_pred,   // [nrows, 100] int64
    const unsigned long long* __restrict__ y_true,   // [nrows, 50]  int64
    const float* __restrict__ multiplier,            // [50] = 1/(j+1)
    float* __restrict__ partials,                    // [gridDim.x] block sums
    float* __restrict__ out,                         // scalar (atomic fallback)
    int nrows)
{
    __shared__ uint32_t shTrue[ROWS_PER_BLOCK][64];  // low dwords of int64s
    __shared__ uint32_t shPred[ROWS_PER_BLOCK][64];
    __shared__ float    shScore[ROWS_PER_BLOCK];

    const int lane = threadIdx.x & 31;
    const int w    = threadIdx.x >> 5;

    int row = blockIdx.x * ROWS_PER_BLOCK + w;
    const bool valid = (row < nrows);
    if (!valid) row = nrows - 1;

    // Element pointers (point at low dword of each little-endian int64).
    const unsigned long long* tRow = y_true + (size_t)row * KTOP;
    const unsigned long long* pRow = y_pred + (size_t)row * PRED_COLS;

    const int ti0 = lane;
    const int ti1 = (lane + 32 < KTOP) ? (lane + 32) : (KTOP - 1); // clamp: no OOB on last row
    const int pi0 = lane;
    const int pi1 = lane + 32;                                      // always < 100

    const uint32_t lT0 = lds_off(&shTrue[w][lane]);
    const uint32_t lT1 = lds_off(&shTrue[w][lane + 32]);
    const uint32_t lP0 = lds_off(&shPred[w][lane]);
    const uint32_t lP1 = lds_off(&shPred[w][lane + 32]);

    // CDNA5 async global->LDS copies (one b32 per lane per instruction);
    // each wave stages its own row slices, tracked by ASYNCcnt.
    asm volatile(
        "global_load_async_to_lds_b32 %0, %4, off\n\t"
        "global_load_async_to_lds_b32 %1, %5, off\n\t"
        "global_load_async_to_lds_b32 %2, %6, off\n\t"
        "global_load_async_to_lds_b32 %3, %7, off"
        :
        : "v"(lT0), "v"(lT1), "v"(lP0), "v"(lP1),
          "v"(tRow + ti0), "v"(tRow + ti1), "v"(pRow + pi0), "v"(pRow + pi1)
        : "memory");
    asm volatile("s_wait_asynccnt 0" ::: "memory");

    // Load the (coalesced, L2-resident) multipliers up front, guarded index.
    const bool  act1  = (lane + 32) < KTOP;            // lanes 0..17 own j = lane+32
    const float mult0 = multiplier[lane];
    const float mult1 = multiplier[act1 ? (lane + 32) : (KTOP - 1)];

    // Each lane evaluates its prediction positions j0 = lane, j1 = lane + 32.
    const uint32_t p0 = shPred[w][lane];
    const uint32_t p1 = shPred[w][lane + 32];

    bool m0 = false, m1 = false;            // appears in y_true?
    #pragma unroll
    for (int t = 0; t < KTOP; ++t) {
        const uint32_t tv = shTrue[w][t];   // broadcast LDS read
        m0 |= (p0 == tv);
        m1 |= (p1 == tv);
    }

    bool d0 = false, d1 = false;            // duplicate of an earlier pred?
    #pragma unroll
    for (int i = 0; i < KTOP - 1; ++i) {
        const uint32_t pv = shPred[w][i];   // broadcast LDS read
        d0 |= (i < lane)      && (p0 == pv);
        d1 |= (i < lane + 32) && (p1 == pv);
    }

    const bool hit0 = valid && m0 && !d0;
    const bool hit1 = valid && act1 && m1 && !d1;

    // Wave-wide hit bitmasks replace the sequential cumsum: prefix popcount.
    const uint32_t maskLo = __builtin_amdgcn_ballot_w32(hit0);  // j = 0..31
    const uint32_t maskHi = __builtin_amdgcn_ballot_w32(hit1);  // j = 32..49
    const uint32_t le     = 0xffffffffu >> (31 - lane);         // bits 0..lane

    float c = 0.0f;
    if (hit0)
        c += (float)__builtin_popcount(maskLo & le) * mult0;
    if (hit1)
        c += (float)(__builtin_popcount(maskLo) +
                     __builtin_popcount(maskHi & le)) * mult1;

    // Wave32 butterfly reduction -> per-row raw score on every lane.
    #pragma unroll
    for (int off = 16; off > 0; off >>= 1)
        c += __shfl_xor(c, off, 32);

    if (lane == 0) shScore[w] = c;
    __syncthreads();

    if (threadIdx.x == 0) {
        float s = 0.0f;
        #pragma unroll
        for (int i = 0; i < ROWS_PER_BLOCK; ++i) s += shScore[i];
        if (USE_ATOMIC) {
            const float scale = (1.0f / (float)KTOP) / (float)nrows;
            atomicAdd(out, s * scale);
        } else {
            partials[blockIdx.x] = s;      // raw block sum; scaled in stage 2
        }
    }
}

// Stage 2: fixed-order reduction of per-block partials -> bit-deterministic.
__global__ __launch_bounds__(256) void mapk_reduce(
    const float* __restrict__ partials, float* __restrict__ out,
    int nparts, int nrows)
{
    __shared__ float sh[256];
    float s = 0.0f;
    for (int i = threadIdx.x; i < nparts; i += 256)  // fixed serial order
        s += partials[i];
    sh[threadIdx.x] = s;
    __syncthreads();
    #pragma unroll
    for (int off = 128; off > 0; off >>= 1) {        // fixed tree order
        if (threadIdx.x < off) sh[threadIdx.x] += sh[threadIdx.x + off];
        __syncthreads();
    }
    if (threadIdx.x == 0) {
        const float scale = (1.0f / (float)KTOP) / (float)nrows;
        out[0] = sh[0] * scale;   // overwrite: no memset needed
    }
}

extern "C" void kernel_launch(void* const* d_in, const int* in_sizes, int n_in,
                              void* d_out, int out_size, void* d_ws, size_t ws_size,
                              hipStream_t stream) {
    const unsigned long long* y_pred = (const unsigned long long*)d_in[0]; // int64
    const unsigned long long* y_true = (const unsigned long long*)d_in[1]; // int64
    const float*              mult   = (const float*)d_in[2];              // f32[50]
    float* out = (float*)d_out;

    const int nrows  = in_sizes[0] / PRED_COLS;   // 16384
    const int blocks = (nrows + ROWS_PER_BLOCK - 1) / ROWS_PER_BLOCK;

    if (d_ws != nullptr && ws_size >= (size_t)blocks * sizeof(float)) {
        // Deterministic path: per-block partials in workspace, fixed-order sum.
        float* partials = (float*)d_ws;
        mapk_kernel<false><<<blocks, 256, 0, stream>>>(
            y_pred, y_true, mult, partials, out, nrows);
        mapk_reduce<<<1, 256, 0, stream>>>(partials, out, blocks, nrows);
    } else {
        // Fallback: atomic accumulation into zeroed scalar.
        hipMemsetAsync(out, 0, sizeof(float), stream);
        mapk_kernel<true><<<blocks, 256, 0, stream>>>(
            y_pred, y_true, mult, nullptr, out, nrows);
    }
}